// RegionLoss_5325759447473
// MI455X (gfx1250) — compile-verified
//
#include <hip/hip_runtime.h>
#include <math.h>

// ---------------- problem constants ----------------
#define NB 64
#define NA 5
#define NC 80
#define NH 32
#define NW 32
#define NT 50
#define PLANE (NH * NW)            // 1024
#define CHS (5 + NC)               // 85
#define BSTRIDE (NA * CHS * PLANE) // 435200
#define PGT (NB * NT)              // 3200

typedef __attribute__((ext_vector_type(2))) float v2f;
typedef __attribute__((ext_vector_type(8))) float v8f;

// ---------------- wave32 sum reduction via V_WMMA_F32_16X16X4_F32 ----------
// A(16x4): VGPR0 lanes0-15 -> K=0 (A[m][0]=v[m]), lanes16-31 -> K=2
// (A[m][2]=v[16+m]); VGPR1 zero. B(4x16) = ones. Then D[m][n] = v[m]+v[16+m]
// for every n. Per lane, the 8 D VGPRs cover M=0..7 (lanes<16) or M=8..15
// (lanes>=16); summing them and adding the lane^16 partner yields the full
// 32-lane sum broadcast to all lanes.
__device__ __forceinline__ float wave_reduce_sum(float v) {
#if __has_builtin(__builtin_amdgcn_wmma_f32_16x16x4_f32)
  v2f A;
  A[0] = v;
  A[1] = 0.0f;
  v2f B;
  B[0] = 1.0f;
  B[1] = 1.0f;
  v8f C = {0.f, 0.f, 0.f, 0.f, 0.f, 0.f, 0.f, 0.f};
  v8f D = __builtin_amdgcn_wmma_f32_16x16x4_f32(false, A, false, B, (short)0,
                                                C, false, false);
  float s = ((D[0] + D[1]) + (D[2] + D[3])) + ((D[4] + D[5]) + (D[6] + D[7]));
  s += __shfl_xor(s, 16, 32);
  return s;
#else
  for (int off = 16; off > 0; off >>= 1)
    v += __shfl_xor(v, off, 32);
  return v;
#endif
}

// ---------------- phase A: per-GT preprocessing (64 threads) ---------------
// ws layout (floats), P = PGT:
//  [0P) gx [1P) gy [2P) gw [3P) gh [4P) valid [5P) cls [6P) tw_val [7P) th_val
//  [8P) best_n (int) [9P) gi (int) [10P) gj (int) [11P) block partials (320)
__global__ void region_prep_kernel(const float* __restrict__ tgt,
                                   float* __restrict__ ws) {
  int b = blockIdx.x * blockDim.x + threadIdx.x;
  if (b >= NB)
    return;
  const float aw[NA] = {1.3221f, 3.19275f, 5.05587f, 9.47112f, 11.2364f};
  const float ah[NA] = {1.73145f, 4.00944f, 8.09892f, 4.84053f, 10.0071f};
  float* gxA = ws;
  float* gyA = ws + PGT;
  float* gwA = ws + 2 * PGT;
  float* ghA = ws + 3 * PGT;
  float* vaA = ws + 4 * PGT;
  float* clA = ws + 5 * PGT;
  float* twA = ws + 6 * PGT;
  float* thA = ws + 7 * PGT;
  int* bnA = (int*)(ws + 8 * PGT);
  int* giA = (int*)(ws + 9 * PGT);
  int* gjA = (int*)(ws + 10 * PGT);

  float run = 1.0f; // cumprod of (gx_norm != 0)
  for (int t = 0; t < NT; ++t) {
    const float* r = tgt + ((size_t)b * NT + t) * 5;
    float cls = r[0];
    float gxn = r[1], gyn = r[2], gwn = r[3], ghn = r[4];
    run *= (gxn != 0.0f) ? 1.0f : 0.0f;
    float gx = gxn * (float)NW, gy = gyn * (float)NH;
    float gw = gwn * (float)NW, gh = ghn * (float)NH;
    // best anchor: centered-box IoU == min(gw,aw)*min(gh,ah)/union
    int bn = 0;
    float best = -1.0f;
    for (int a = 0; a < NA; ++a) {
      float iw = fminf(gw, aw[a]);
      float ih = fminf(gh, ah[a]);
      float inter = (iw <= 0.0f || ih <= 0.0f) ? 0.0f : iw * ih;
      float uni = gw * gh + aw[a] * ah[a] - inter;
      float iou = inter / uni;
      if (iou > best) { // strict > keeps first max (jnp.argmax tie rule)
        best = iou;
        bn = a;
      }
    }
    int gi = (int)floorf(gx);
    gi = gi < 0 ? 0 : (gi > NW - 1 ? NW - 1 : gi);
    int gj = (int)floorf(gy);
    gj = gj < 0 ? 0 : (gj > NH - 1 ? NH - 1 : gj);

    int idx = b * NT + t;
    gxA[idx] = gx;
    gyA[idx] = gy;
    gwA[idx] = gw;
    ghA[idx] = gh;
    vaA[idx] = run;
    clA[idx] = cls;
    twA[idx] = (run > 0.0f) ? logf(gw / aw[bn]) : 0.0f;
    thA[idx] = (run > 0.0f) ? logf(gh / ah[bn]) : 0.0f;
    bnA[idx] = bn;
    giA[idx] = gi;
    gjA[idx] = gj;
  }
}

// ---------------- phase B: main loss, one block per (b, a) -----------------
__global__ void region_loss_kernel(const float* __restrict__ out,
                                   const float* __restrict__ gws,
                                   float* __restrict__ partials) {
  __shared__ float s_gx[NT], s_gy[NT], s_gw[NT], s_gh[NT];
  __shared__ float s_va[NT], s_cl[NT], s_tw[NT], s_th[NT];
  __shared__ int s_bn[NT], s_gi[NT], s_gj[NT];
  __shared__ float s_red[8];

  const int blk = blockIdx.x;
  const int b = blk / NA;
  const int a = blk % NA;
  const int tid = threadIdx.x;

  if (tid < NT) {
    int idx = b * NT + tid;
    s_gx[tid] = gws[idx];
    s_gy[tid] = gws[PGT + idx];
    s_gw[tid] = gws[2 * PGT + idx];
    s_gh[tid] = gws[3 * PGT + idx];
    s_va[tid] = gws[4 * PGT + idx];
    s_cl[tid] = gws[5 * PGT + idx];
    s_tw[tid] = gws[6 * PGT + idx];
    s_th[tid] = gws[7 * PGT + idx];
    s_bn[tid] = ((const int*)(gws + 8 * PGT))[idx];
    s_gi[tid] = ((const int*)(gws + 9 * PGT))[idx];
    s_gj[tid] = ((const int*)(gws + 10 * PGT))[idx];
  }
  __syncthreads();

  const float anchW[NA] = {1.3221f, 3.19275f, 5.05587f, 9.47112f, 11.2364f};
  const float anchH[NA] = {1.73145f, 4.00944f, 8.09892f, 4.84053f, 10.0071f};
  const float* base = out + (size_t)b * BSTRIDE + (size_t)a * CHS * PLANE;

  float part = 0.0f;
  for (int cell = tid; cell < PLANE; cell += blockDim.x) {
    const int j = cell >> 5;
    const int i = cell & 31;
    const float* p = base + cell;
    const float xv = p[0];
    const float yv = p[PLANE];
    const float wv = p[2 * PLANE];
    const float hv = p[3 * PLANE];
    const float cv = p[4 * PLANE];
    const float sx = 1.0f / (1.0f + expf(-xv));
    const float sy = 1.0f / (1.0f + expf(-yv));
    const float sc = 1.0f / (1.0f + expf(-cv));
    const float px = sx + (float)i;
    const float py = sy + (float)j;
    const float pw = expf(wv) * anchW[a];
    const float ph = expf(hv) * anchH[a];

    float cur = 0.0f, hit = 0.0f;
    float txa = 0.0f, tya = 0.0f, twa = 0.0f, tha = 0.0f;
    float tca = 0.0f, tcl = 0.0f;
    for (int t = 0; t < NT; ++t) {
      const float va = s_va[t];
      const float gx = s_gx[t], gy = s_gy[t], gw = s_gw[t], gh = s_gh[t];
      const float mx = fminf(px - pw * 0.5f, gx - gw * 0.5f);
      const float Mx = fmaxf(px + pw * 0.5f, gx + gw * 0.5f);
      const float my = fminf(py - ph * 0.5f, gy - gh * 0.5f);
      const float My = fmaxf(py + ph * 0.5f, gy + gh * 0.5f);
      const float cw = pw + gw - (Mx - mx);
      const float chh = ph + gh - (My - my);
      const float inter = (cw <= 0.0f || chh <= 0.0f) ? 0.0f : cw * chh;
      const float uni = pw * ph + gw * gh - inter;
      const float iou = inter / uni;
      cur = fmaxf(cur, iou * va);
      // gather form of the reference scatter: this cell is pred_sel for t
      if (s_bn[t] == a && s_gi[t] == i && s_gj[t] == j) {
        hit += va;
        txa += va * (gx - (float)s_gi[t]);
        tya += va * (gy - (float)s_gj[t]);
        twa += va * s_tw[t];
        tha += va * s_th[t];
        tca += va * iou; // iou_gt == IoU(gt, pred at this cell)
        tcl += va * s_cl[t];
      }
    }
    const bool hb = hit > 0.0f;
    const float tx = hb ? txa : 0.5f;
    const float ty = hb ? tya : 0.5f;
    const float tw = hb ? twa : 0.0f;
    const float th = hb ? tha : 0.0f;
    const float tconf = hb ? tca : 0.0f;
    const float cmask = hb ? 5.0f : ((cur > 0.6f) ? 0.0f : 1.0f);
    const float dx = sx - tx, dy = sy - ty;
    const float dw = wv - tw, dh = hv - th, dc = sc - tconf;
    part += 0.5f * (dx * dx + dy * dy + dw * dw + dh * dh) +
            0.5f * cmask * dc * dc;
    if (hb) { // class CE only at hit cells (<1% of cells)
      int ci = (int)tcl;
      ci = ci < 0 ? 0 : (ci > NC - 1 ? NC - 1 : ci);
      const float* cl = p + 5 * PLANE;
      float m = cl[0];
      for (int c = 1; c < NC; ++c)
        m = fmaxf(m, cl[c * PLANE]);
      float sum = 0.0f;
      for (int c = 0; c < NC; ++c)
        sum += expf(cl[c * PLANE] - m);
      part += (m + logf(sum)) - cl[ci * PLANE];
    }
  }

  // block reduction: WMMA wave sums -> LDS -> WMMA again in wave 0
  const float wsum = wave_reduce_sum(part);
  const int lane = tid & 31;
  const int wid = tid >> 5;
  if (lane == 0)
    s_red[wid] = wsum;
  __syncthreads();
  if (wid == 0) {
    const float v = (lane < (int)(blockDim.x >> 5)) ? s_red[lane] : 0.0f;
    const float tot = wave_reduce_sum(v);
    if (lane == 0)
      partials[blk] = tot;
  }
}

// ---------------- phase C: deterministic final reduction (1 wave) ----------
__global__ void region_final_kernel(const float* __restrict__ partials,
                                    float* __restrict__ outv, int n) {
  const int lane = threadIdx.x;
  float s = 0.0f;
  for (int k = lane; k < n; k += 32)
    s += partials[k];
  s = wave_reduce_sum(s);
  if (lane == 0)
    outv[0] = s;
}

extern "C" void kernel_launch(void* const* d_in, const int* in_sizes, int n_in,
                              void* d_out, int out_size, void* d_ws,
                              size_t ws_size, hipStream_t stream) {
  (void)in_sizes;
  (void)n_in;
  (void)out_size;
  (void)ws_size;
  const float* output = (const float*)d_in[0];
  const float* target = (const float*)d_in[1];
  float* ws = (float*)d_ws;
  float* partials = ws + 11 * PGT; // 320 floats of block partials

  region_prep_kernel<<<1, 64, 0, stream>>>(target, ws);
  region_loss_kernel<<<NB * NA, 256, 0, stream>>>(output, ws, partials);
  region_final_kernel<<<1, 32, 0, stream>>>(partials, (float*)d_out, NB * NA);
}